// FrequentDirections_4999341932646
// MI455X (gfx1250) — compile-verified
//
#include <hip/hip_runtime.h>

typedef __attribute__((ext_vector_type(2))) float v2f;
typedef __attribute__((ext_vector_type(8))) float v8f;
typedef __attribute__((ext_vector_type(4))) unsigned int u32x4;
typedef __attribute__((ext_vector_type(8))) unsigned int u32x8;

#define NDIM 128      // feature dim D
#define LTOK 64       // sketch rows l
#define GSTR 130      // G/V row stride (pad: 130 mod 64 = 2)
#define XSTR 132      // X staging row stride (pad: 132 mod 64 = 4) == TDM pad 4 dw / 128 dw
#define NSWEEP 10     // fixed Jacobi sweeps (deterministic)

// Issue a TDM DMA: 64x128 fp32 row-major tile, global -> LDS, with 4-DWORD
// row padding every 128 DWORDs (reproduces the stride-132 LDS layout).
// D# layout per CDNA5 ISA.
__device__ __forceinline__ void tdm_load_chunk(const float* src, float* dstLds) {
    u32x4 g0;
    g0[0] = 1u;                                   // count=1 (valid), user mode
    g0[1] = (unsigned int)(uintptr_t)dstLds;      // lds_addr (flat low 32 bits = LDS offset)
    unsigned long long ga = (unsigned long long)(uintptr_t)src;
    g0[2] = (unsigned int)ga;                     // global_addr[31:0]
    g0[3] = ((unsigned int)(ga >> 32) & 0x01FFFFFFu) | (2u << 30); // addr[56:32] | type=2

    u32x8 g1;
    g1[0] = (2u << 16)        // data_size = 2 -> 4 bytes
          | (1u << 20)        // pad_enable
          | (6u << 22)        // pad_interval enc 6 -> every 128 DWORDs
          | (3u << 25);       // pad_amount  enc 3 -> 4 DWORDs
    g1[1] = (128u << 16);     // tensor_dim0 = 128 elems (bits 79:48)
    g1[2] = (64u  << 16);     // tensor_dim1 = 64 rows  (bits 111:80)
    g1[3] = (128u << 16);     // tile_dim0   = 128      (bits 127:112)
    g1[4] = 64u;              // tile_dim1   = 64       (bits 143:128); tile_dim2 = 0
    g1[5] = 128u;             // tensor_dim0_stride = 128 (bits 207:160)
    g1[6] = 0u;
    g1[7] = 0u;

    u32x4 gz = {0u, 0u, 0u, 0u};                  // groups 2/3 unused (2-D tile)
    asm volatile("tensor_load_to_lds %0, %1, %2, %3"
                 :: "s"(g0), "s"(g1), "s"(gz), "s"(gz)
                 : "memory");
}

// One block per (b,h). 256 threads = 8 waves (wave32).
// Stage 1: G = X^T X via V_WMMA_F32_16X16X4_F32; X streamed by TDM (double buffered).
// Stage 2: parallel cyclic Jacobi eigendecomposition of G in LDS.
// Stage 3: rank eigenvalues, shrink spectrum, write B = diag(s_new) * Vh[:64].
__global__ __launch_bounds__(256) void fd_sketch_kernel(
    const float* __restrict__ kv,      // [BH, 1024, 128]
    const float* __restrict__ sketch,  // [BH,   64, 128]
    float* __restrict__ out)           // [BH,   64, 128]
{
    extern __shared__ float smem[];
    float* G  = smem;                          // 128*130 floats (stage1: staging buf A)
    float* V  = smem + NDIM * GSTR;            // 128*130 floats (stage1: staging buf B)
    float* XcA = G;                            // 64*132 floats <= 128*130
    float* XcB = V;
    float* rc = smem + 2 * NDIM * GSTR;        // 64 rotation cosines
    float* rs = rc + 64;                       // 64 rotation sines
    int*   rp = (int*)(rs + 64);               // 64 pair p
    int*   rq = rp + 64;                       // 64 pair q
    int*   inv = rq + 64;                      // 64 rank -> eig index
    float* sv  = (float*)(inv + 64);           // 64 shrunk singular values
    float* lam64 = sv + 64;                    // 1: eigenvalue at rank 64

    const int tid  = threadIdx.x;
    const int lane = tid & 31;
    const int wave = tid >> 5;                 // 0..7
    const int bh   = blockIdx.x;               // 0..255

    const float* kvp = kv     + (size_t)bh * 1024 * NDIM;
    const float* skp = sketch + (size_t)bh * LTOK * NDIM;

    // ---------------- Stage 1: Gram matrix via WMMA + TDM pipeline ----------------
    v8f zero8 = {};
    v8f acc[8];
    #pragma unroll
    for (int j = 0; j < 8; ++j) acc[j] = zero8;

    const int i0    = wave * 16;               // this wave's row block of G
    const int l15   = lane & 15;
    const int kbase = (lane < 16) ? 0 : 2;     // f32 A/B fragment K split across lane halves

    if (wave == 0) {                           // prefetch chunk 0 (sketch) into buf A
        tdm_load_chunk(skp, XcA);
    }

    for (int chunk = 0; chunk < 17; ++chunk) { // 1 sketch chunk + 16 kv chunks of 64 rows
        float* cur = (chunk & 1) ? XcB : XcA;
        if (wave == 0) {
            if (chunk + 1 < 17) {              // prefetch next chunk into other buffer
                float* nxt = (chunk & 1) ? XcA : XcB;
                tdm_load_chunk(kvp + (size_t)chunk * 64 * NDIM, nxt);
                __builtin_amdgcn_s_wait_tensorcnt(1);  // in-order: chunk c has landed
            } else {
                __builtin_amdgcn_s_wait_tensorcnt(0);
            }
        }
        __syncthreads();                       // publish chunk c to all waves
        #pragma unroll
        for (int ks = 0; ks < 16; ++ks) {      // 64 rows / K=4 per WMMA
            const int k0 = ks * 4 + kbase;
            v2f a;                             // A = X^T tile: A[m][k] = X[k0+k][i0+m]
            a.x = cur[k0 * XSTR + i0 + l15];
            a.y = cur[(k0 + 1) * XSTR + i0 + l15];
            #pragma unroll
            for (int j = 0; j < 8; ++j) {
                v2f b;                         // B[k][n] = X[k0+k][16j+n]
                b.x = cur[k0 * XSTR + j * 16 + l15];
                b.y = cur[(k0 + 1) * XSTR + j * 16 + l15];
                acc[j] = __builtin_amdgcn_wmma_f32_16x16x4_f32(
                    false, a, false, b, (short)0, acc[j], false, false);
            }
        }
        __syncthreads();                       // chunk fully consumed; buffer reusable
    }

    // Spill accumulators to G (C/D layout: VGPR v -> M = v + 8*(lane>=16), N = lane&15)
    {
        const int mAdd = (lane < 16) ? 0 : 8;
        #pragma unroll
        for (int j = 0; j < 8; ++j) {
            #pragma unroll
            for (int v = 0; v < 8; ++v) {
                G[(i0 + v + mAdd) * GSTR + j * 16 + l15] = acc[j][v];
            }
        }
    }
    // V = I (overwrites staging; safe after final sync above)
    for (int i = tid; i < NDIM * GSTR; i += 256) V[i] = 0.0f;
    __syncthreads();
    if (tid < NDIM) V[tid * GSTR + tid] = 1.0f;
    __syncthreads();

    // ---------------- Stage 2: parallel cyclic Jacobi ----------------
    for (int sweep = 0; sweep < NSWEEP; ++sweep) {
        for (int r = 0; r < 127; ++r) {
            if (tid < 64) {
                int p, q;
                if (tid == 0) { p = 127; q = r; }
                else { p = (r + tid) % 127; q = (r + 127 - tid) % 127; }
                if (p > q) { int t = p; p = q; q = t; }
                float app = G[p * GSTR + p];
                float aqq = G[q * GSTR + q];
                float apq = G[p * GSTR + q];
                float c = 1.0f, s = 0.0f;
                if (fabsf(apq) > 1e-20f) {
                    float tau = (aqq - app) / (2.0f * apq);
                    float t = (tau >= 0.0f ? 1.0f : -1.0f)
                              / (fabsf(tau) + sqrtf(1.0f + tau * tau));
                    c = rsqrtf(1.0f + t * t);
                    s = t * c;
                }
                rc[tid] = c; rs[tid] = s; rp[tid] = p; rq[tid] = q;
            }
            __syncthreads();
            // Row update: G <- J^T G  (disjoint row pairs)
            #pragma unroll 4
            for (int i = 0; i < 32; ++i) {
                int idx = tid + i * 256;         // 64 pairs x 128 cols
                int k = idx >> 7, col = idx & 127;
                int p = rp[k], q = rq[k];
                float c = rc[k], s = rs[k];
                float gp = G[p * GSTR + col], gq = G[q * GSTR + col];
                G[p * GSTR + col] = c * gp - s * gq;
                G[q * GSTR + col] = s * gp + c * gq;
            }
            __syncthreads();
            // Column update: G <- G J, V <- V J  (disjoint column pairs)
            #pragma unroll 4
            for (int i = 0; i < 32; ++i) {
                int idx = tid + i * 256;         // 64 pairs x 128 rows
                int k = idx >> 7, row = idx & 127;
                int p = rp[k], q = rq[k];
                float c = rc[k], s = rs[k];
                float gp = G[row * GSTR + p], gq = G[row * GSTR + q];
                G[row * GSTR + p] = c * gp - s * gq;
                G[row * GSTR + q] = s * gp + c * gq;
                float vp = V[row * GSTR + p], vq = V[row * GSTR + q];
                V[row * GSTR + p] = c * vp - s * vq;
                V[row * GSTR + q] = s * vp + c * vq;
            }
            __syncthreads();
        }
    }

    // ---------------- Stage 3: rank, shrink, emit ----------------
    if (tid < NDIM) {
        float lam = G[tid * GSTR + tid];
        int rank = 0;
        for (int j = 0; j < NDIM; ++j) {
            float lj = G[j * GSTR + j];
            rank += (lj > lam) || (lj == lam && j < tid);
        }
        if (rank == LTOK) lam64[0] = lam;          // delta = sigma_{l}^2
        if (rank < LTOK)  inv[rank] = tid;
    }
    __syncthreads();
    if (tid < LTOK) {
        int i = inv[tid];
        float d = G[i * GSTR + i] - lam64[0];
        sv[tid] = sqrtf(d > 0.0f ? d : 0.0f);
    }
    __syncthreads();

    float* outp = out + (size_t)bh * LTOK * NDIM;
    #pragma unroll 4
    for (int it = 0; it < 32; ++it) {
        int idx = tid + it * 256;                   // 64 rows x 128 cols
        int row = idx >> 7, d = idx & 127;
        int i = inv[row];
        outp[row * NDIM + d] = sv[row] * V[d * GSTR + i];  // Vh row = eigencolumn^T
    }
}

extern "C" void kernel_launch(void* const* d_in, const int* in_sizes, int n_in,
                              void* d_out, int out_size, void* d_ws, size_t ws_size,
                              hipStream_t stream) {
    (void)in_sizes; (void)n_in; (void)out_size; (void)d_ws; (void)ws_size;
    const float* kv     = (const float*)d_in[0];  // [8,32,1024,128] f32
    const float* sketch = (const float*)d_in[1];  // [8,32,64,128]   f32
    float* out = (float*)d_out;                   // [8,32,64,128]   f32
    // G + V (each 128*130 floats) + 1024 floats of small scratch
    size_t smem_bytes = (size_t)(2 * NDIM * GSTR + 1024) * sizeof(float);
    fd_sketch_kernel<<<dim3(8 * 32), dim3(256), smem_bytes, stream>>>(kv, sketch, out);
}